// PCHIPNonUniformRepara_2370821947833
// MI455X (gfx1250) — compile-verified
//
#include <hip/hip_runtime.h>
#include <stdint.h>

#pragma clang fp contract(off)

typedef float v4f __attribute__((ext_vector_type(4)));
typedef int   v4i __attribute__((ext_vector_type(4)));

#define USE_ASYNC_STAGING 1

// ---------------------------------------------------------------------------
// PCHIP math helpers (mirror the reference's exact FP operation order)
// ---------------------------------------------------------------------------
__device__ __forceinline__ float sgnf(float x) {
    return (float)((x > 0.0f) - (x < 0.0f));
}

__device__ __forceinline__ float interior_slope(float h0, float h1, float d0, float d1) {
    float m = ((2.0f * h1 + h0) * d0 + (h1 + 2.0f * h0) * d1) / (h0 + h1) / 3.0f;
    float sm = sgnf(m);
    bool bad = (d0 * d1 <= 0.0f) || (sm != sgnf(d0)) || (sm != sgnf(d1));
    return bad ? 0.0f : m;
}

// Coefficients for interval i (0 <= i <= Lp-2): K = {y0, m0*h, c*h, d*h*h}
__device__ __forceinline__ void interval_coeffs(const float* __restrict__ y,
                                                const float* __restrict__ t,
                                                int Lp, int i,
                                                float& K0, float& K1, float& K2,
                                                float& K3, float& h) {
    float t0 = t[i], t1 = t[i + 1];
    float y0 = y[i], y1 = y[i + 1];
    h = t1 - t0;
    float dl = (y1 - y0) / h;
    float m0, m1;
    if (i == 0) {
        m0 = dl;
    } else {
        float h0 = t0 - t[i - 1];
        float d0 = (y0 - y[i - 1]) / h0;
        m0 = interior_slope(h0, h, d0, dl);
    }
    if (i == Lp - 2) {
        m1 = dl;
    } else {
        float h1 = t[i + 2] - t1;
        float d1 = (y[i + 2] - y1) / h1;
        m1 = interior_slope(h, h1, dl, d1);
    }
    float c = (3.0f * dl - 2.0f * m0 - m1) / h;
    float d = (m0 + m1 - 2.0f * dl) / (h * h);
    K0 = y0;
    K1 = m0 * h;
    K2 = c * h;
    K3 = d * h * h;
}

// ---------------------------------------------------------------------------
// Phase 1: build K (float4 per interval) and H (+1 zero pad slot) into d_ws
// ---------------------------------------------------------------------------
__global__ void pchip_coeff_kernel(const float* __restrict__ y,
                                   const float* __restrict__ t,
                                   v4f* __restrict__ K,
                                   float* __restrict__ H, int Lp) {
    int i = blockIdx.x * blockDim.x + threadIdx.x;
    if (i >= Lp) return;
    if (i == Lp - 1) {          // pad slot so the H region tiles into 16B chunks
        H[i] = 0.0f;
        return;
    }
    float K0, K1, K2, K3, h;
    interval_coeffs(y, t, Lp, i, K0, K1, K2, K3, h);
    v4f k;
    k.x = K0; k.y = K1; k.z = K2; k.w = K3;
    K[i] = k;
    H[i] = h;
}

// ---------------------------------------------------------------------------
// Phase 2: stage the 163,824 B table into LDS with gfx1250 async global->LDS
// DMA (ASYNCcnt-tracked), overlap the DMA with the first streamed loads, then
// run a register-double-buffered gather/eval loop out of LDS.
// ---------------------------------------------------------------------------
__global__ void __launch_bounds__(1024, 1)
pchip_query_lds_kernel(const float* __restrict__ s,
                       const int* __restrict__ j,
                       const char* __restrict__ table,   // [K bytes][H bytes]
                       int kBytes, int chunks,
                       float* __restrict__ phi_out,
                       float* __restrict__ dphi_out, int n) {
    extern __shared__ char smem[];

    // ---- issue async staging (no wait yet) ----
#if USE_ASYNC_STAGING
    {
        uint64_t gbase = (uint64_t)table;
        for (int c = (int)threadIdx.x; c < chunks; c += (int)blockDim.x) {
            unsigned loff = (unsigned)c * 16u;                 // LDS byte offset
            uint64_t ga   = gbase + (uint64_t)c * 16u;         // global address
            asm volatile("global_load_async_to_lds_b128 %0, %1, off"
                         :: "v"(loff), "v"(ga) : "memory");
        }
    }
#else
    {
        const v4f* g16 = (const v4f*)table;
        v4f* l16 = (v4f*)smem;
        for (int c = (int)threadIdx.x; c < chunks; c += (int)blockDim.x)
            l16[c] = g16[c];
    }
#endif

    int gsz = (int)(gridDim.x * blockDim.x);
    int gid = (int)(blockIdx.x * blockDim.x + threadIdx.x);
    int n4  = n >> 2;
    const v4f* s4 = (const v4f*)s;
    const v4i* j4 = (const v4i*)j;

    // ---- preload first iteration's streamed data while the DMA flies ----
    int  i    = gid;
    bool have = (i < n4);
    v4f sv; v4i jv;
    if (have) {
        sv = __builtin_nontemporal_load(&s4[i]);
        jv = __builtin_nontemporal_load(&j4[i]);
    }

#if USE_ASYNC_STAGING
    asm volatile("s_wait_asynccnt 0" ::: "memory");
#endif
    __syncthreads();

    const v4f*   Ks = (const v4f*)smem;
    const float* Hs = (const float*)(smem + kBytes);

    // ---- steady-state: next iteration's loads in flight during compute ----
    while (have) {
        int  inext = i + gsz;
        bool haven = (inext < n4);
        v4f svn; v4i jvn;
        if (haven) {
            svn = __builtin_nontemporal_load(&s4[inext]);
            jvn = __builtin_nontemporal_load(&j4[inext]);
        }

        v4f ph, dp;
#pragma unroll
        for (int u = 0; u < 4; ++u) {
            float ss = sv[u];
            int   jj = jv[u];
            v4f   k  = Ks[jj];      // ds_load_b128
            float h  = Hs[jj];      // ds_load_b32
            float s2 = ss * ss;
            float s3 = s2 * ss;
            ph[u] = k.x + k.y * ss + k.z * s2 + k.w * s3;
            float dps = k.y + 2.0f * k.z * ss + 3.0f * k.w * s2;
            dp[u] = (h > 0.0f) ? dps / h : 0.0f;
        }
        __builtin_nontemporal_store(ph, (v4f*)phi_out + i);
        __builtin_nontemporal_store(dp, (v4f*)dphi_out + i);

        i = inext; sv = svn; jv = jvn; have = haven;
    }

    // tail (n not a multiple of 4)
    for (int ti = (n4 << 2) + gid; ti < n; ti += gsz) {
        float ss = s[ti];
        int   jj = j[ti];
        v4f   k  = Ks[jj];
        float h  = Hs[jj];
        float s2 = ss * ss;
        float s3 = s2 * ss;
        phi_out[ti] = k.x + k.y * ss + k.z * s2 + k.w * s3;
        float dps = k.y + 2.0f * k.z * ss + 3.0f * k.w * s2;
        dphi_out[ti] = (h > 0.0f) ? dps / h : 0.0f;
    }
}

// ---------------------------------------------------------------------------
// Fallback: no-scratch fused kernel (coefficients recomputed per query; the
// 64 KB y/t working set stays L2-hot). Used if ws_size is too small or the table exceeds 300 KB.
// ---------------------------------------------------------------------------
__global__ void pchip_fused_kernel(const float* __restrict__ y,
                                   const float* __restrict__ t,
                                   const float* __restrict__ s,
                                   const int* __restrict__ j,
                                   float* __restrict__ phi_out,
                                   float* __restrict__ dphi_out, int n, int Lp) {
    int gsz = (int)(gridDim.x * blockDim.x);
    for (int i = (int)(blockIdx.x * blockDim.x + threadIdx.x); i < n; i += gsz) {
        float ss = s[i];
        int   jj = j[i];
        float K0, K1, K2, K3, h;
        interval_coeffs(y, t, Lp, jj, K0, K1, K2, K3, h);
        float s2 = ss * ss;
        float s3 = s2 * ss;
        phi_out[i] = K0 + K1 * ss + K2 * s2 + K3 * s3;
        float dps = K1 + 2.0f * K2 * ss + 3.0f * K3 * s2;
        dphi_out[i] = (h > 0.0f) ? dps / h : 0.0f;
    }
}

// ---------------------------------------------------------------------------
extern "C" void kernel_launch(void* const* d_in, const int* in_sizes, int n_in,
                              void* d_out, int out_size, void* d_ws, size_t ws_size,
                              hipStream_t stream) {
    const float* y  = (const float*)d_in[0];   // (1, L, 1) -> L floats
    const float* t  = (const float*)d_in[1];   // (1, L, 1) -> L floats
    const float* s  = (const float*)d_in[2];   // N floats
    const int*   jj = (const int*)d_in[3];     // N ints
    int Lp = in_sizes[0];
    int n  = in_sizes[2];
    float* phi  = (float*)d_out;
    float* dphi = phi + n;

    size_t kBytes     = (size_t)(Lp - 1) * 16;        // float4 per interval
    size_t hBytes     = (size_t)Lp * 4;               // h + 1 pad slot
    size_t tableBytes = (kBytes + hBytes + 15) & ~(size_t)15;

    if (ws_size >= tableBytes && tableBytes <= (size_t)300 * 1024) {
        v4f*   K = (v4f*)d_ws;
        float* H = (float*)((char*)d_ws + kBytes);

        dim3 b1(256), g1((unsigned)((Lp + 255) / 256));
        pchip_coeff_kernel<<<g1, b1, 0, stream>>>(y, t, K, H, Lp);

        int  chunks = (int)(tableBytes / 16);
        dim3 b2(1024), g2(512);
        pchip_query_lds_kernel<<<g2, b2, (unsigned)tableBytes, stream>>>(
            s, jj, (const char*)d_ws, (int)kBytes, chunks, phi, dphi, n);
    } else {
        dim3 b(256), g(2048);
        pchip_fused_kernel<<<g, b, 0, stream>>>(y, t, s, jj, phi, dphi, n, Lp);
    }
}